// HyperbolicAttention_80418967650991
// MI455X (gfx1250) — compile-verified
//
#include <hip/hip_runtime.h>
#include <math.h>

typedef float v2f __attribute__((ext_vector_type(2)));
typedef float v8f __attribute__((ext_vector_type(8)));

#define B_ROWS 8192
#define DIM 256
#define EPSF 1e-7f
#define M_TILE 16
#define N_TILE 16
#define THREADS 256
#define NKT (B_ROWS / N_TILE)

// ---------------------------------------------------------------------------
// CDNA5 async cache->LDS copy (16B per lane), tracked by ASYNCcnt.
// ---------------------------------------------------------------------------
__device__ __forceinline__ void async_copy16(unsigned lds_off, const float* g) {
  asm volatile("global_load_async_to_lds_b128 %0, %1, off"
               :
               : "v"(lds_off), "v"(g)
               : "memory");
}
__device__ __forceinline__ void wait_async0() {   // all copies done
  asm volatile("s_wait_asynccnt 0x0" ::: "memory");
}
__device__ __forceinline__ void wait_async8() {   // oldest tile (8 copies) done
  asm volatile("s_wait_asynccnt 0x8" ::: "memory");
}

// ---------------------------------------------------------------------------
// Kernel 1: row squared-norms for Q and K  ->  uu[8192], vv[8192] in d_ws
// ---------------------------------------------------------------------------
__global__ __launch_bounds__(256) void hyp_norms(const float* __restrict__ q,
                                                 const float* __restrict__ k,
                                                 float* __restrict__ uu,
                                                 float* __restrict__ vv) {
  int gw = (blockIdx.x * blockDim.x + threadIdx.x) >> 5;  // one wave per row
  int lane = threadIdx.x & 31;
  const float* src = (gw < B_ROWS) ? q : k;
  float* dst = (gw < B_ROWS) ? uu : vv;
  int row = (gw < B_ROWS) ? gw : gw - B_ROWS;
  const float* p = src + (size_t)row * DIM;
  float s = 0.f;
#pragma unroll
  for (int i = 0; i < DIM / 32; ++i) {
    float x = p[lane + i * 32];
    s += x * x;
  }
#pragma unroll
  for (int off = 16; off > 0; off >>= 1) s += __shfl_xor(s, off, 32);
  if (lane == 0) dst[row] = s;
}

// ---------------------------------------------------------------------------
// Kernel 2: fused hyperbolic flash-attention, double-buffered async staging.
// Block = 256 threads (8 waves), owns M_TILE=16 query rows.
// Wave w owns output columns [w*32, w*32+32) and the k-slice [w*32, w*32+32)
// of the split-K QK^T. Elementwise distance/softmax is distributed: each
// thread owns score element (row = 2w+h, key = n); running stats in LDS.
// Key/value tiles for iteration t+1 stream cache->LDS (ASYNCcnt) behind the
// WMMA + softmax work of iteration t.
// ---------------------------------------------------------------------------
__global__ __launch_bounds__(256) void hyp_attn(const float* __restrict__ Q,
                                                const float* __restrict__ Km,
                                                const float* __restrict__ Vm,
                                                const float* __restrict__ cptr,
                                                const float* __restrict__ uu,
                                                const float* __restrict__ vv,
                                                float* __restrict__ out) {
  __shared__ __align__(16) float sK[2][16][260];  // K tiles (padded, transposed B reads)
  __shared__ __align__(16) float sV[2][16][264];  // V tiles (padded, halves in disjoint banks)
  __shared__ float sPart[8][16][18];              // split-K partial score tiles
  __shared__ float sP[16][20];                    // exp'd probabilities (A-layout source)
  __shared__ float sM[16];                        // running row max of (-dist)
  __shared__ float sL[16];                        // running softmax denominator
  __shared__ float sScale[16];                    // per-iteration accumulator rescale

  const int tid = threadIdx.x;
  const int w = tid >> 5;      // wave id: k-split slot, partial slot, col group
  const int lane = tid & 31;
  const int h = lane >> 4;     // half-wave
  const int n = lane & 15;     // N / M index inside fragments
  const int row0 = blockIdx.x * M_TILE;
  const int c0 = w * 32;       // this wave's output column base
  const int srow = 2 * w + h;  // softmax row owned by this half-wave
  const int kb = w * 32;       // this wave's K-dim slice

  // curvature preprocessing (matches reference clipping)
  const float c_cl = fminf(fmaxf(fabsf(cptr[0]) + EPSF, EPSF), 1.f - EPSF);
  const float inv_csqrt = rsqrtf(c_cl);

  if (tid < 16) { sM[tid] = -1e30f; sL[tid] = 0.f; }

  // stage a 16-key K/V tile into buffer b (8 async b128 copies per wave)
  auto stage = [&](int key0, int b) {
#pragma unroll
    for (int j = 0; j < 4; ++j) {
      int i = tid + j * THREADS;
      int r = i >> 6, c4 = i & 63;
      async_copy16((unsigned)(size_t)&sK[b][r][c4 * 4],
                   Km + (size_t)(key0 + r) * DIM + c4 * 4);
      async_copy16((unsigned)(size_t)&sV[b][r][c4 * 4],
                   Vm + (size_t)(key0 + r) * DIM + c4 * 4);
    }
  };

  // Preload loop-invariant QK^T A-fragments (this wave's rows x k-slice)
  // straight from global (one-time; Q is L2 resident).
  v2f aq[8];
#pragma unroll
  for (int ks4 = 0; ks4 < 8; ++ks4) {
    int kk = kb + ks4 * 4 + 2 * h;
    aq[ks4] = *(const v2f*)(Q + (size_t)(row0 + n) * DIM + kk);
  }

  // query-side hyperbolic terms for the softmax row this thread owns
  const float uur = uu[row0 + srow];
  const float omu = 1.f - fminf(c_cl * uur, 1.f - EPSF);

  v8f o0 = {0.f, 0.f, 0.f, 0.f, 0.f, 0.f, 0.f, 0.f};
  v8f o1 = {0.f, 0.f, 0.f, 0.f, 0.f, 0.f, 0.f, 0.f};

  stage(0, 0);  // prologue: tile 0 in flight

  for (int kt = 0; kt < NKT; ++kt) {
    const int b = kt & 1;
    const int key0 = kt * N_TILE;
    __syncthreads();  // bar1: previous iteration done reading buffer b^1

    if (kt + 1 < NKT) {
      stage(key0 + N_TILE, b ^ 1);  // prefetch next tile behind this iteration
      wait_async8();                // oldest 8 copies (= tile kt) have landed
    } else {
      wait_async0();
    }
    __syncthreads();  // bar2: tile kt resident in buffer b for all waves

    // ---- split-K QK^T: this wave covers k in [w*32, w*32+32) ----
    v8f acc = {0.f, 0.f, 0.f, 0.f, 0.f, 0.f, 0.f, 0.f};
#pragma unroll
    for (int ks4 = 0; ks4 < 8; ++ks4) {
      int kk = kb + ks4 * 4 + 2 * h;
      v2f bfrag = *(const v2f*)&sK[b][n][kk];  // B: N(key)=n, K = {kk, kk+1}
      acc = __builtin_amdgcn_wmma_f32_16x16x4_f32(false, aq[ks4], false, bfrag,
                                                  (short)0, acc, false, false);
    }
#pragma unroll
    for (int r = 0; r < 8; ++r) sPart[w][r + 8 * h][n] = acc[r];
    __syncthreads();  // bar3: partials visible

    // ---- distributed reduce + hyperbolic distance + online softmax ----
    {
      float s = 0.f;
#pragma unroll
      for (int pw = 0; pw < 8; ++pw) s += sPart[pw][srow][n];
      float vvj = vv[key0 + n];
      float omv = 1.f - fminf(c_cl * vvj, 1.f - EPSF);
      float diff = fmaxf(uur + vvj - 2.f * s, 0.f);
      float denom = fmaxf(omu * omv, EPSF);
      float arg = fmaxf(1.f + 2.f * c_cl * diff / denom, 1.f + EPSF);
      float dist = __logf(arg + __fsqrt_rn(arg * arg - 1.f)) * inv_csqrt;  // acosh
      float nd = -dist;
      float rmax = nd;  // row max across the 16 lanes of this half-wave
      rmax = fmaxf(rmax, __shfl_xor(rmax, 1, 32));
      rmax = fmaxf(rmax, __shfl_xor(rmax, 2, 32));
      rmax = fmaxf(rmax, __shfl_xor(rmax, 4, 32));
      rmax = fmaxf(rmax, __shfl_xor(rmax, 8, 32));
      float mold = sM[srow];
      float mnew = fmaxf(mold, rmax);
      float scl = __expf(mold - mnew);
      float p = __expf(nd - mnew);
      float rs = p;
      rs += __shfl_xor(rs, 1, 32);
      rs += __shfl_xor(rs, 2, 32);
      rs += __shfl_xor(rs, 4, 32);
      rs += __shfl_xor(rs, 8, 32);
      if (n == 0) {
        sM[srow] = mnew;
        sL[srow] = sL[srow] * scl + rs;
        sScale[srow] = scl;
      }
      sP[srow][n] = p;
    }
    __syncthreads();  // bar4: sP / sScale visible

    // ---- rescale accumulators by this iteration's row factors ----
#pragma unroll
    for (int r = 0; r < 8; ++r) {
      float scl = sScale[r + 8 * h];
      o0[r] *= scl;
      o1[r] *= scl;
    }

    // ---- P(16x16) x V(16 x 32-col slice) ----
#pragma unroll
    for (int ks = 0; ks < 16; ks += 4) {
      int kk = ks + 2 * h;
      v2f a = *(const v2f*)&sP[n][kk];  // A: M=n, K(key) = {kk, kk+1}
      v2f b0, b1;
      b0.x = sV[b][kk][c0 + n];       b0.y = sV[b][kk + 1][c0 + n];
      b1.x = sV[b][kk][c0 + 16 + n];  b1.y = sV[b][kk + 1][c0 + 16 + n];
      o0 = __builtin_amdgcn_wmma_f32_16x16x4_f32(false, a, false, b0,
                                                 (short)0, o0, false, false);
      o1 = __builtin_amdgcn_wmma_f32_16x16x4_f32(false, a, false, b1,
                                                 (short)0, o1, false, false);
    }
  }
  __syncthreads();

  // ---- epilogue: normalize by softmax denominator, write out ----
#pragma unroll
  for (int r = 0; r < 8; ++r) {
    float inv = 1.f / sL[r + 8 * h];
    size_t orow = (size_t)(row0 + r + 8 * h) * DIM;
    out[orow + c0 + n] = o0[r] * inv;
    out[orow + c0 + 16 + n] = o1[r] * inv;
  }
}

// ---------------------------------------------------------------------------
extern "C" void kernel_launch(void* const* d_in, const int* in_sizes, int n_in,
                              void* d_out, int out_size, void* d_ws, size_t ws_size,
                              hipStream_t stream) {
  (void)in_sizes; (void)n_in; (void)out_size; (void)ws_size;
  const float* Q = (const float*)d_in[0];
  const float* K = (const float*)d_in[1];
  const float* V = (const float*)d_in[2];
  const float* c = (const float*)d_in[3];
  float* out = (float*)d_out;
  float* uu = (float*)d_ws;   // 8192 floats
  float* vv = uu + B_ROWS;    // 8192 floats

  hyp_norms<<<(2 * B_ROWS) / 8, THREADS, 0, stream>>>(Q, K, uu, vv);
  hyp_attn<<<B_ROWS / M_TILE, THREADS, 0, stream>>>(Q, K, V, c, uu, vv, out);
}